// GGNN_62732292325694
// MI455X (gfx1250) — compile-verified
//
#include <hip/hip_runtime.h>
#include <hip/hip_bf16.h>
#include <math.h>

// ---------------- problem constants ----------------
#define NN      100000      // nodes
#define NE      800000      // edges
#define NG      64          // graphs
#define NT      4           // edge types
#define DD      150         // feature dim
#define DP      160         // padded feature dim (mult of 16, K mult of 4)
#define G3      450         // 3*D gates
#define GP      480         // padded gates (mult of 16)
#define HID     512
#define NPASS   5

typedef __attribute__((ext_vector_type(2))) float v2f;
typedef __attribute__((ext_vector_type(8))) float v8f;

// ---------------- WMMA FP32 GEMM ----------------
// One wave computes one 16x16 tile of Out = A[M,K] * B[K,N] (+addSrc)(+colBias).
// A row-major lda, B row-major ldb, Out row-major ldo. K % 4 == 0, N tiles via gridDim.y.
// FP32 WMMA (16x16x4) keeps full reference precision; workload is scatter-bound
// so the lower fp32 matrix rate is immaterial.
#define GEMM_WAVES 4
__global__ __launch_bounds__(GEMM_WAVES * 32)
void gemm_wmma_f32(const float* __restrict__ A, int lda,
                   const float* __restrict__ B, int ldb,
                   const float* __restrict__ addSrc,
                   const float* __restrict__ colBias,
                   float* __restrict__ Out, int ldo,
                   int M, int K) {
    const int lane  = threadIdx.x & 31;
    const int tileM = blockIdx.x * GEMM_WAVES + (threadIdx.x >> 5);
    if (tileM * 16 >= M) return;              // wave-uniform: EXEC stays all-1s
    const int mr    = lane & 15;              // row within tile (A) / col within tile (B)
    const int khalf = (lane >> 4) << 1;       // 0 or 2: which K pair this lane holds

    const float* __restrict__ Arow = A + (size_t)(tileM * 16 + mr) * (size_t)lda + khalf;
    const float* __restrict__ Bcol = B + (size_t)khalf * (size_t)ldb + blockIdx.y * 16 + mr;

    v8f acc = {};
    for (int k0 = 0; k0 < K; k0 += 4) {
        v2f a, b;
        a[0] = Arow[k0];
        a[1] = Arow[k0 + 1];
        b[0] = Bcol[(size_t)k0 * (size_t)ldb];
        b[1] = Bcol[(size_t)(k0 + 1) * (size_t)ldb];
        acc = __builtin_amdgcn_wmma_f32_16x16x4_f32(
            false, a, false, b, (short)0, acc, false, false);
    }

    const int mbase = tileM * 16 + ((lane >> 4) << 3);   // M = r + 8*(lane>=16)
    const int col   = blockIdx.y * 16 + mr;
    const float bv  = colBias ? colBias[col] : 0.0f;
#pragma unroll
    for (int r = 0; r < 8; ++r) {
        size_t o = (size_t)(mbase + r) * (size_t)ldo + col;
        float v = acc[r] + bv;
        if (addSrc) v += addSrc[o];
        Out[o] = v;
    }
}

// ---------------- prep kernels ----------------
// Pack/transpose weights into padded K-major layouts:
//  Wt[t][d][o]  = W_e[t][o][d]   (4 x 160 x 160)
//  Wih[d][g]    = w_ih[g][d]     (160 x 480)
//  Whh[d][g]    = w_hh[g][d]     (160 x 480)
//  bihp/bhhp    = padded biases  (480)
__global__ void prep_weights(const float* __restrict__ W_e,
                             const float* __restrict__ w_ih,
                             const float* __restrict__ w_hh,
                             const float* __restrict__ b_ih,
                             const float* __restrict__ b_hh,
                             float* __restrict__ Wt,
                             float* __restrict__ Wih,
                             float* __restrict__ Whh,
                             float* __restrict__ bihp,
                             float* __restrict__ bhhp) {
    int gid = blockIdx.x * blockDim.x + threadIdx.x;
    if (gid < NT * DP * DP) {
        int t = gid / (DP * DP);
        int d = (gid / DP) % DP;
        int o = gid % DP;
        float v = 0.0f;
        if (d < DD && o < DD) v = W_e[(size_t)t * DD * DD + (size_t)o * DD + d];
        Wt[gid] = v;
        return;
    }
    int g2 = gid - NT * DP * DP;
    if (g2 < DP * GP) {
        int d = g2 / GP, g = g2 % GP;
        float v = 0.0f;
        if (d < DD && g < G3) v = w_ih[(size_t)g * DD + d];
        Wih[g2] = v;
        return;
    }
    int g3 = g2 - DP * GP;
    if (g3 < DP * GP) {
        int d = g3 / GP, g = g3 % GP;
        float v = 0.0f;
        if (d < DD && g < G3) v = w_hh[(size_t)g * DD + d];
        Whh[g3] = v;
        return;
    }
    int g4 = g3 - DP * GP;
    if (g4 < GP) { bihp[g4] = (g4 < G3) ? b_ih[g4] : 0.0f; return; }
    int g5 = g4 - GP;
    if (g5 < GP) { bhhp[g5] = (g5 < G3) ? b_hh[g5] : 0.0f; return; }
}

// h[n][0:150] = nodes, pad cols 150..159 = 0
__global__ void init_h(const float* __restrict__ nodes, float* __restrict__ h) {
    int idx = blockIdx.x * blockDim.x + threadIdx.x;   // over NN*DP
    int n = idx / DP, d = idx % DP;
    h[idx] = (d < DD) ? nodes[(size_t)n * DD + d] : 0.0f;
}

// per-node per-type in-degree (float so a single atomic op type suffices)
__global__ void deg_count(const int* __restrict__ edst, const int* __restrict__ etyp,
                          float* __restrict__ deg) {
    int e = blockIdx.x * blockDim.x + threadIdx.x;
    if (e >= NE) return;
    atomicAdd(&deg[(size_t)edst[e] * NT + etyp[e]], 1.0f);
}

// bias_inc[n][d] = sum_t deg[n][t] * b_e[t][d]  (padded cols zero)
__global__ void bias_inc_kernel(const float* __restrict__ deg,
                                const float* __restrict__ b_e,
                                float* __restrict__ bias_inc) {
    int idx = blockIdx.x * blockDim.x + threadIdx.x;   // over NN*DP
    int n = idx / DP, d = idx % DP;
    float v = 0.0f;
    if (d < DD) {
#pragma unroll
        for (int t = 0; t < NT; ++t)
            v += deg[(size_t)n * NT + t] * b_e[(size_t)t * DD + d];
    }
    bias_inc[idx] = v;
}

// scatter edges of one type: agg[dst] += h[src]  (one wave per edge)
__global__ void scatter_edges(const int* __restrict__ esrc,
                              const int* __restrict__ edst,
                              const int* __restrict__ etyp, int t,
                              const float* __restrict__ h,
                              float* __restrict__ agg) {
    int gtid = blockIdx.x * blockDim.x + threadIdx.x;
    int e    = gtid >> 5;
    int lane = gtid & 31;
    if (e >= NE) return;
    if (etyp[e] != t) return;
    const float* __restrict__ src = h + (size_t)esrc[e] * DP;
    float* __restrict__ dst = agg + (size_t)edst[e] * DP;
    for (int d = lane; d < DD; d += 32)
        atomicAdd(&dst[d], src[d]);
}

__device__ __forceinline__ float sigmoidf_(float x) { return 1.0f / (1.0f + expf(-x)); }

// GRU gates, in-place h update (element-local, safe)
__global__ void gru_update(const float* __restrict__ gi,
                           const float* __restrict__ gh,
                           float* __restrict__ h) {
    int idx = blockIdx.x * blockDim.x + threadIdx.x;   // over NN*DD
    if (idx >= NN * DD) return;
    int n = idx / DD, d = idx % DD;
    const float* __restrict__ gin = gi + (size_t)n * GP;
    const float* __restrict__ ghn = gh + (size_t)n * GP;
    float r  = sigmoidf_(gin[d]        + ghn[d]);
    float z  = sigmoidf_(gin[DD + d]   + ghn[DD + d]);
    float nn = tanhf(gin[2 * DD + d] + r * ghn[2 * DD + d]);
    float* hp = h + (size_t)n * DP + d;
    *hp = (1.0f - z) * nn + z * (*hp);
}

// graph readout sum
__global__ void readout_sum(const float* __restrict__ h,
                            const int* __restrict__ gids,
                            float* __restrict__ gsum) {
    int idx = blockIdx.x * blockDim.x + threadIdx.x;   // over NN*DD
    if (idx >= NN * DD) return;
    int n = idx / DD, d = idx % DD;
    atomicAdd(&gsum[(size_t)gids[n] * DD + d], h[(size_t)n * DP + d]);
}

// final tiny MLP: one block per graph
__global__ void readout_mlp(const float* __restrict__ gsum,
                            const float* __restrict__ pclass,
                            const float* __restrict__ fc1w,
                            const float* __restrict__ fc1b,
                            const float* __restrict__ fc2w,
                            const float* __restrict__ fc2b,
                            float* __restrict__ out) {
    __shared__ float l[DD + 1];
    __shared__ float hid[HID];
    int b = blockIdx.x, tid = threadIdx.x;
    for (int i = tid; i < DD; i += blockDim.x) {
        float v = logf(gsum[(size_t)b * DD + i]);
        if (v != v) v = 0.0f;          // NaN -> 0
        l[i] = fmaxf(v, 0.0f);         // relu (also kills -inf from log(0))
    }
    if (tid == 0) l[DD] = pclass[b];
    __syncthreads();
    for (int j = tid; j < HID; j += blockDim.x) {
        float s = fc1b[j];
        for (int i = 0; i < DD + 1; ++i) s += l[i] * fc1w[(size_t)i * HID + j];
        hid[j] = (s > 0.0f) ? s : 0.01f * s;   // leaky_relu
    }
    __syncthreads();
    for (int o = tid; o < 10; o += blockDim.x) {
        float s = fc2b[o];
        for (int i = 0; i < HID; ++i) s += hid[i] * fc2w[(size_t)i * 10 + o];
        out[(size_t)b * 10 + o] = s;
    }
}

// ---------------- launch ----------------
extern "C" void kernel_launch(void* const* d_in, const int* in_sizes, int n_in,
                              void* d_out, int out_size, void* d_ws, size_t ws_size,
                              hipStream_t stream) {
    (void)in_sizes; (void)n_in; (void)out_size; (void)ws_size;
    const float* nodes  = (const float*)d_in[0];
    const float* pclass = (const float*)d_in[1];
    const int*   esrc   = (const int*)d_in[2];
    const int*   edst   = (const int*)d_in[3];
    const int*   etyp   = (const int*)d_in[4];
    const int*   gids   = (const int*)d_in[5];
    const float* W_e    = (const float*)d_in[6];
    const float* b_e    = (const float*)d_in[7];
    const float* w_ih   = (const float*)d_in[8];
    const float* w_hh   = (const float*)d_in[9];
    const float* b_ih   = (const float*)d_in[10];
    const float* b_hh   = (const float*)d_in[11];
    const float* fc1w   = (const float*)d_in[12];
    const float* fc1b   = (const float*)d_in[13];
    const float* fc2w   = (const float*)d_in[14];
    const float* fc2b   = (const float*)d_in[15];
    float* outp = (float*)d_out;

    // workspace carve-up (all offsets 256B aligned)
    char* ws = (char*)d_ws;
    size_t off = 0;
    float* h        = (float*)(ws + off); off += (size_t)NN * DP * 4;        // 64 MB
    float* agg      = (float*)(ws + off); off += (size_t)NN * DP * 4;        // 64 MB
    float* inc      = (float*)(ws + off); off += (size_t)NN * DP * 4;        // 64 MB
    float* bias_inc = (float*)(ws + off); off += (size_t)NN * DP * 4;        // 64 MB
    float* gi       = (float*)(ws + off); off += (size_t)NN * GP * 4;        // 192 MB
    float* gh       = (float*)(ws + off); off += (size_t)NN * GP * 4;        // 192 MB
    float* deg      = (float*)(ws + off); off += (size_t)NN * NT * 4;        // 1.6 MB
    float* Wt       = (float*)(ws + off); off += (size_t)NT * DP * DP * 4;
    float* Wih      = (float*)(ws + off); off += (size_t)DP * GP * 4;
    float* Whh      = (float*)(ws + off); off += (size_t)DP * GP * 4;
    float* bihp     = (float*)(ws + off); off += 2048;
    float* bhhp     = (float*)(ws + off); off += 2048;
    float* gsum     = (float*)(ws + off); off += (size_t)NG * DD * 4;

    const int prepN = NT * DP * DP + 2 * DP * GP + 2 * GP;

    hipMemsetAsync(deg, 0, (size_t)NN * NT * 4, stream);
    prep_weights<<<(prepN + 255) / 256, 256, 0, stream>>>(
        W_e, w_ih, w_hh, b_ih, b_hh, Wt, Wih, Whh, bihp, bhhp);
    init_h<<<(NN * DP) / 256, 256, 0, stream>>>(nodes, h);
    deg_count<<<NE / 256, 256, 0, stream>>>(edst, etyp, deg);
    bias_inc_kernel<<<(NN * DP) / 256, 256, 0, stream>>>(deg, b_e, bias_inc);

    const int mTiles  = NN / 16;                                   // 6250
    const int gemmGx  = (mTiles + GEMM_WAVES - 1) / GEMM_WAVES;    // 1563

    for (int pass = 0; pass < NPASS; ++pass) {
        for (int t = 0; t < NT; ++t) {
            hipMemsetAsync(agg, 0, (size_t)NN * DP * 4, stream);
            scatter_edges<<<(NE * 32) / 256, 256, 0, stream>>>(esrc, edst, etyp, t, h, agg);
            // inc = (t==0 ? bias_inc : inc) + agg @ W_e[t]^T
            gemm_wmma_f32<<<dim3(gemmGx, DP / 16), GEMM_WAVES * 32, 0, stream>>>(
                agg, DP, Wt + (size_t)t * DP * DP, DP,
                (t == 0) ? bias_inc : inc, nullptr, inc, DP, NN, DP);
        }
        // gi = inc @ w_ih^T + b_ih ; gh = h @ w_hh^T + b_hh
        gemm_wmma_f32<<<dim3(gemmGx, GP / 16), GEMM_WAVES * 32, 0, stream>>>(
            inc, DP, Wih, GP, nullptr, bihp, gi, GP, NN, DP);
        gemm_wmma_f32<<<dim3(gemmGx, GP / 16), GEMM_WAVES * 32, 0, stream>>>(
            h, DP, Whh, GP, nullptr, bhhp, gh, GP, NN, DP);
        gru_update<<<(NN * DD + 255) / 256, 256, 0, stream>>>(gi, gh, h);
    }

    hipMemsetAsync(gsum, 0, (size_t)NG * DD * 4, stream);
    readout_sum<<<(NN * DD + 255) / 256, 256, 0, stream>>>(h, gids, gsum);
    readout_mlp<<<NG, 256, 0, stream>>>(gsum, pclass, fc1w, fc1b, fc2w, fc2b, outp);
}